// EXPPos_20684562497573
// MI455X (gfx1250) — compile-verified
//
#include <hip/hip_runtime.h>

// CDNA5 / gfx1250: wave32. Streaming elementwise 16-step spiking recurrence.
// Divide eliminated: (v-T)/(|v|+1) > 0  <=>  v > T (denominator > 0).
// Two elements packed per state pair -> per-step updates use V_PK_FMA_F32
// (2 FMAs/lane/instr). z in {0,1} keeps all products exact (bitwise == ref);
// negated state w = -v is exact (RN(-a) = -RN(a)).
// 16 elements/thread to amortize the per-thread preamble (s_loads, pair
// materialization) over more payload.

typedef float v4f __attribute__((ext_vector_type(4)));
typedef float v2f __attribute__((ext_vector_type(2)));

__device__ __forceinline__ v2f exppos_pair(float x0, float x1,
                                           const float hs[16],
                                           const float ds[16],
                                           const float nTs[16]) {
    v2f w;
    w.x = __int_as_float(__float_as_int(x0) | 0x80000000);  // -|x0|
    w.y = __int_as_float(__float_as_int(x1) | 0x80000000);  // -|x1|

    // Peeled step 0 (z_prev = 0 -> no w-update): z = (v > T0) <=> (w < -T0)
    v2f zz;
    zz.x = (w.x < nTs[0]) ? 1.0f : 0.0f;
    zz.y = (w.y < nTs[0]) ? 1.0f : 0.0f;
    v2f dd0 = {ds[0], ds[0]};
    v2f ac  = zz * dd0;                       // v_pk_mul_f32

#pragma unroll
    for (int t = 1; t < 16; ++t) {
        v2f hh = {hs[t], hs[t]};              // loop-invariant splat
        w = __builtin_elementwise_fma(zz, hh, w);   // v_pk_fma_f32: w += z*h
        zz.x = (w.x < nTs[t]) ? 1.0f : 0.0f;        // spike per element
        zz.y = (w.y < nTs[t]) ? 1.0f : 0.0f;
        v2f dd = {ds[t], ds[t]};
        ac = __builtin_elementwise_fma(zz, dd, ac); // v_pk_fma_f32: acc += z*d
    }

    // out * sign(x): acc >= 0 (all d[t] > 0) -> copysign + exact-zero select.
    float r0 = __builtin_copysignf(ac.x, x0);
    float r1 = __builtin_copysignf(ac.y, x1);
    v2f r;
    r.x = (x0 != 0.0f) ? r0 : 0.0f;
    r.y = (x1 != 0.0f) ? r1 : 0.0f;
    return r;
}

__device__ __forceinline__ float exppos_one(float x,
                                            const float hs[16],
                                            const float ds[16],
                                            const float nTs[16]) {
    float w   = __int_as_float(__float_as_int(x) | 0x80000000);
    float z   = (w < nTs[0]) ? 1.0f : 0.0f;
    float acc = z * ds[0];
#pragma unroll
    for (int t = 1; t < 16; ++t) {
        w   = fmaf(z, hs[t], w);
        z   = (w < nTs[t]) ? 1.0f : 0.0f;
        acc = fmaf(z, ds[t], acc);
    }
    float r = __builtin_copysignf(acc, x);
    return (x != 0.0f) ? r : 0.0f;
}

__global__ __launch_bounds__(256) void exppos_v16_kernel(
    const v4f* __restrict__ x4,
    const float* __restrict__ h,
    const float* __restrict__ d,
    const float* __restrict__ T,
    v4f* __restrict__ out4,
    int n16)   // number of 16-element groups
{
    int idx = blockIdx.x * 256 + threadIdx.x;
    if (idx >= n16) return;

    // Lane-invariant addresses -> SMEM s_load_b512; params live in SGPRs.
    float hs[16], ds[16], nTs[16];
#pragma unroll
    for (int t = 0; t < 16; ++t) {
        hs[t]  = h[t];
        ds[t]  = d[t];
        nTs[t] = -T[t];
    }

    // Streaming 512MB >> 192MB L2: non-temporal b128 loads/stores (s_clause'd).
    v4f a = __builtin_nontemporal_load(x4 + 4 * idx);
    v4f b = __builtin_nontemporal_load(x4 + 4 * idx + 1);
    v4f c = __builtin_nontemporal_load(x4 + 4 * idx + 2);
    v4f e = __builtin_nontemporal_load(x4 + 4 * idx + 3);

    // 8 packed pairs = 16 elements, each pair driven by v_pk_fma_f32.
    v2f p0 = exppos_pair(a[0], a[1], hs, ds, nTs);
    v2f p1 = exppos_pair(a[2], a[3], hs, ds, nTs);
    v2f p2 = exppos_pair(b[0], b[1], hs, ds, nTs);
    v2f p3 = exppos_pair(b[2], b[3], hs, ds, nTs);
    v2f p4 = exppos_pair(c[0], c[1], hs, ds, nTs);
    v2f p5 = exppos_pair(c[2], c[3], hs, ds, nTs);
    v2f p6 = exppos_pair(e[0], e[1], hs, ds, nTs);
    v2f p7 = exppos_pair(e[2], e[3], hs, ds, nTs);

    v4f oa, ob, oc, oe;
    oa[0] = p0.x; oa[1] = p0.y; oa[2] = p1.x; oa[3] = p1.y;
    ob[0] = p2.x; ob[1] = p2.y; ob[2] = p3.x; ob[3] = p3.y;
    oc[0] = p4.x; oc[1] = p4.y; oc[2] = p5.x; oc[3] = p5.y;
    oe[0] = p6.x; oe[1] = p6.y; oe[2] = p7.x; oe[3] = p7.y;

    __builtin_nontemporal_store(oa, out4 + 4 * idx);
    __builtin_nontemporal_store(ob, out4 + 4 * idx + 1);
    __builtin_nontemporal_store(oc, out4 + 4 * idx + 2);
    __builtin_nontemporal_store(oe, out4 + 4 * idx + 3);
}

__global__ __launch_bounds__(256) void exppos_tail_kernel(
    const float* __restrict__ x,
    const float* __restrict__ h,
    const float* __restrict__ d,
    const float* __restrict__ T,
    float* __restrict__ out,
    int start, int n)
{
    int i = start + blockIdx.x * 256 + threadIdx.x;
    if (i >= n) return;
    float hs[16], ds[16], nTs[16];
#pragma unroll
    for (int t = 0; t < 16; ++t) { hs[t] = h[t]; ds[t] = d[t]; nTs[t] = -T[t]; }
    out[i] = exppos_one(x[i], hs, ds, nTs);
}

extern "C" void kernel_launch(void* const* d_in, const int* in_sizes, int n_in,
                              void* d_out, int out_size, void* d_ws, size_t ws_size,
                              hipStream_t stream) {
    const float* x = (const float*)d_in[0];
    const float* h = (const float*)d_in[1];
    const float* d = (const float*)d_in[2];
    const float* T = (const float*)d_in[3];
    float* out = (float*)d_out;

    const int n   = in_sizes[0];      // 8*2048*4096 = 67,108,864
    const int n16 = n >> 4;           // 16-element groups (4,194,304)
    const int rem_start = n16 << 4;
    const int rem = n - rem_start;

    if (n16 > 0) {
        int grid = (n16 + 255) / 256; // 16384 blocks
        exppos_v16_kernel<<<grid, 256, 0, stream>>>(
            (const v4f*)x, h, d, T, (v4f*)out, n16);
    }
    if (rem > 0) {
        int grid = (rem + 255) / 256;
        exppos_tail_kernel<<<grid, 256, 0, stream>>>(
            x, h, d, T, out, rem_start, n);
    }
}